// Resample2d_62388694942048
// MI455X (gfx1250) — compile-verified
//
#include <hip/hip_runtime.h>
#include <stdint.h>

#define HH 1080
#define WW 1920
#define NN (HH * WW)          // 2,073,600 ; divisible by 256 (8100 blocks)
#define BLK 256

typedef unsigned int v4u __attribute__((ext_vector_type(4)));
typedef int          v8i __attribute__((ext_vector_type(8)));
typedef int          v4i __attribute__((ext_vector_type(4)));

// Issue one Tensor Data Mover load: a 1-row tile of `BLK` elements
// (element size = 1<<dsz_log bytes) from global `gptr` into LDS `laddr`.
// tensor_dim0 = remaining elements so the DMA's own OOB check zero-fills tails.
__device__ __forceinline__ void tdm_load_row(const void* gptr, uint32_t laddr,
                                             uint32_t dsz_log, uint32_t remaining) {
    uint64_t gaddr = (uint64_t)(uintptr_t)gptr;

    // D# group 0: count=1 | lds_addr | global_addr[56:0] | type=2
    v4u g0;
    g0[0] = 1u;
    g0[1] = laddr;
    g0[2] = (uint32_t)gaddr;
    g0[3] = (uint32_t)((gaddr >> 32) & 0x1FFFFFFu) | (2u << 30);

    // D# group 1: data_size | tensor_dim0=remaining | tensor_dim1=1 |
    //             tile_dim0=BLK | tile_dim1=1 | stride0=remaining
    v8i g1;
    g1[0] = (int)(dsz_log << 16);
    g1[1] = (int)((remaining & 0xFFFFu) << 16);
    g1[2] = (int)(((remaining >> 16) & 0xFFFFu) | (1u << 16));
    g1[3] = (int)((uint32_t)BLK << 16);
    g1[4] = 1;
    g1[5] = (int)remaining;
    g1[6] = 0;
    g1[7] = 0;

    v4i g2 = {0, 0, 0, 0};
    v4i g3 = {0, 0, 0, 0};
#if __clang_major__ >= 23
    v8i g4 = {0, 0, 0, 0, 0, 0, 0, 0};
    __builtin_amdgcn_tensor_load_to_lds(g0, g1, g2, g3, g4, 0);
#else
    __builtin_amdgcn_tensor_load_to_lds(g0, g1, g2, g3, 0);
#endif
}

// --------------------------------------------------------------- splat ------
// First function in the module so the disasm snippet shows the TDM path.
// Z-buffer splat with deterministic 64-bit key:
//   key = depth_bits[31:0] << 32 | source_index  (smaller depth wins;
//   positive floats order-preserve as uints)
__global__ void splat_kernel(const float* __restrict__ flow,
                             const float* __restrict__ depth,
                             const unsigned int* __restrict__ maxslot,
                             unsigned long long* __restrict__ zbuf) {
    __shared__ float2 sflow[BLK];   // 2 KB flow tile
    __shared__ float  sdepth[BLK];  // 1 KB depth tile

    const int base = blockIdx.x * BLK;

    if (threadIdx.x < 32) {  // wave 0 only: execz branch keeps other waves out;
                             // TDM ignores EXEC once issued, so one DMA per block
        uint32_t remaining = (uint32_t)(NN - base);
        tdm_load_row(flow + 2 * (size_t)base, (uint32_t)(uintptr_t)&sflow[0],
                     /*8B*/ 3u, remaining);
        tdm_load_row(depth + (size_t)base, (uint32_t)(uintptr_t)&sdepth[0],
                     /*4B*/ 2u, remaining);
        __builtin_amdgcn_s_wait_tensorcnt(0);
    }
    __syncthreads();

    const int i = base + threadIdx.x;           // NN % BLK == 0: no tail
    const float2 f = sflow[threadIdx.x];
    const float  d = sdepth[threadIdx.x];

    const int y = i / WW;
    const int x = i - y * WW;
    // jnp.round == round-half-to-even == rintf under default RNE mode
    const int tx = (int)__builtin_rintf((float)x + f.x);
    const int ty = (int)__builtin_rintf((float)y + f.y);

    const unsigned int dbits = __float_as_uint(d);
    const unsigned int mb    = *maxslot;  // uniform; global-max pixel excluded
                                          // (reference uses strict `depth < mids[split]`)
    if ((unsigned)tx < (unsigned)WW && (unsigned)ty < (unsigned)HH && dbits != mb) {
        unsigned long long key =
            ((unsigned long long)dbits << 32) | (unsigned int)i;
        atomicMin(&zbuf[(size_t)ty * WW + tx], key);
    }
}

// ---------------------------------------------------------------- init ------
__global__ void init_kernel(ulonglong2* __restrict__ zbuf2,
                            float4* __restrict__ out4,
                            unsigned int* __restrict__ maxslot) {
    int i      = blockIdx.x * blockDim.x + threadIdx.x;
    int stride = gridDim.x * blockDim.x;
    ulonglong2 ones;
    ones.x = ~0ull;
    ones.y = ~0ull;
    for (int k = i; k < NN / 2; k += stride) zbuf2[k] = ones;       // b128 stores
    float4 z = make_float4(0.f, 0.f, 0.f, 0.f);
    for (int k = i; k < 3 * NN / 4; k += stride) out4[k] = z;       // b128 stores
    if (i == 0) *maxslot = 0u;
}

// ------------------------------------------------------------ max depth -----
__global__ void maxdepth_kernel(const float4* __restrict__ depth4,
                                unsigned int* __restrict__ maxslot) {
    int i      = blockIdx.x * blockDim.x + threadIdx.x;
    int stride = gridDim.x * blockDim.x;
    unsigned int m = 0u;
    for (int k = i; k < NN / 4; k += stride) {                      // b128 loads
        __builtin_prefetch(depth4 + k + stride, 0, 1);              // global_prefetch_b8
        float4 v = depth4[k];
        unsigned int b0 = __float_as_uint(v.x), b1 = __float_as_uint(v.y);
        unsigned int b2 = __float_as_uint(v.z), b3 = __float_as_uint(v.w);
        unsigned int p = b0 > b1 ? b0 : b1;
        unsigned int q = b2 > b3 ? b2 : b3;
        unsigned int r = p > q ? p : q;
        m = m > r ? m : r;
    }
    // wave32 butterfly reduction
    for (int off = 16; off > 0; off >>= 1) {
        unsigned int o = (unsigned int)__shfl_xor((int)m, off, 32);
        m = m > o ? m : o;
    }
    if ((threadIdx.x & 31) == 0) atomicMax(maxslot, m);
}

// -------------------------------------------------------------- resolve -----
__global__ void resolve_kernel(const unsigned long long* __restrict__ zbuf,
                               const float* __restrict__ img,
                               float* __restrict__ out) {
    int t = blockIdx.x * blockDim.x + threadIdx.x;
    if (t >= NN) return;
    // z-buffer and output are touched exactly once -> nontemporal
    unsigned long long key = __builtin_nontemporal_load(&zbuf[t]);
    if (key != ~0ull) {
        size_t src = (size_t)(unsigned int)key;
        float r = img[3 * src + 0];
        float g = img[3 * src + 1];
        float b = img[3 * src + 2];
        size_t o = 3 * (size_t)t;
        __builtin_nontemporal_store(r, &out[o + 0]);
        __builtin_nontemporal_store(g, &out[o + 1]);
        __builtin_nontemporal_store(b, &out[o + 2]);
    }
    // misses keep the zeros written by init_kernel (reference: res starts at 0)
}

// ------------------------------------------------------------------ launch --
extern "C" void kernel_launch(void* const* d_in, const int* in_sizes, int n_in,
                              void* d_out, int out_size, void* d_ws, size_t ws_size,
                              hipStream_t stream) {
    (void)in_sizes; (void)n_in; (void)out_size; (void)ws_size;

    const float* img   = (const float*)d_in[0];
    const float* flow  = (const float*)d_in[1];   // [H,W,2] : (dx, dy)
    const float* depth = (const float*)d_in[2];
    // d_in[3] = split: quantile bins collapse under the min-depth z-buffer rule.

    float* out = (float*)d_out;

    unsigned long long* zbuf = (unsigned long long*)d_ws;                  // NN * 8 B
    unsigned int* maxslot    = (unsigned int*)((char*)d_ws + (size_t)NN * 8);

    init_kernel<<<4096, BLK, 0, stream>>>((ulonglong2*)zbuf, (float4*)out, maxslot);
    maxdepth_kernel<<<2048, BLK, 0, stream>>>((const float4*)depth, maxslot);
    splat_kernel<<<NN / BLK, BLK, 0, stream>>>(flow, depth, maxslot, zbuf);
    resolve_kernel<<<NN / BLK, BLK, 0, stream>>>(zbuf, img, out);
}